// IngredientPositionEncoding_84026740178995
// MI455X (gfx1250) — compile-verified
//
#include <hip/hip_runtime.h>
#include <hip/hip_bf16.h>

typedef __attribute__((ext_vector_type(2))) float v2f;
typedef __attribute__((ext_vector_type(8))) float v8f;

#define BB 64
#define LL 4096
#define DD 128
#define MAXSEG 256

// ---------------------------------------------------------------------------
// Kernel 1: per-batch scan of the separator mask.
// Produces, for each segment s in [0, 256):
//   segstart[b*256+s] = first token index of segment s
//   segcnt  [b*256+s] = number of (non-separator) tokens in segment s
// Construction guarantees start[] and end[]=start+cnt are monotone in s,
// so a 16-segment tile's tokens are exactly [start[s0], end[s0+15]).
// ---------------------------------------------------------------------------
__global__ void __launch_bounds__(256)
seg_scan_kernel(const int* __restrict__ mask,
                int* __restrict__ segstart,
                int* __restrict__ segcnt) {
    const int b   = blockIdx.x;
    const int tid = threadIdx.x;                 // 256 threads, 16 tokens each
    __shared__ int scan[256];
    __shared__ int seppos[MAXSEG];

    const int* m = mask + (size_t)b * LL;
    const int base = tid * 16;

    // local separator count
    int c = 0;
    #pragma unroll
    for (int k = 0; k < 16; ++k) c += (m[base + k] != 0);

    seppos[tid] = LL;                            // init (only < nsep entries read)
    scan[tid] = c;
    __syncthreads();
    // Hillis-Steele inclusive scan over 256 entries
    for (int off = 1; off < 256; off <<= 1) {
        int add = (tid >= off) ? scan[tid - off] : 0;
        __syncthreads();
        scan[tid] += add;
        __syncthreads();
    }
    // second pass: write separator positions by global rank
    int e = scan[tid] - c;                       // exclusive rank
    for (int k = 0; k < 16; ++k) {
        if (m[base + k] != 0) {
            if (e < MAXSEG) seppos[e] = base + k;
            ++e;
        }
    }
    __syncthreads();

    const int nsep = scan[255];
    const int s = tid;                           // one segment per thread
    int start, end;
    if (s < nsep) {                              // bounded by separator s
        start = (s == 0) ? 0 : seppos[s - 1] + 1;
        end   = seppos[s];
    } else if (s == nsep) {                      // trailing tokens after last sep
        start = (s == 0) ? 0 : seppos[s - 1] + 1;
        end   = LL;
    } else {                                     // past the data: empty at L
        start = LL;
        end   = LL;
    }
    segstart[b * MAXSEG + s] = start;
    segcnt  [b * MAXSEG + s] = end - start;
}

// ---------------------------------------------------------------------------
// Kernel 2: segment mean via V_WMMA_F32_16X16X4_F32.
// Block = (batch b, segment tile st: segments [16*st, 16*st+16)).
// 8 waves; wave w owns dim tile d0 = 16*w. Each wave accumulates a 16x16 f32
// C tile: C[s,d] += A[s,k] * B[k,d], A = segment-membership selector (1/0),
// B = x[t,d] slices. Main loop: 16 tokens / iteration = one s_clause of 8
// global_load_b32 (immediate offsets) + 4 WMMAs, pointer-bump addressing, no
// clamping. Clamped 4-token tail handles the ragged end.
//
// VGPR layouts per CDNA5 ISA:
//   A (16x4 f32, 2 VGPRs): lane M = lane&15; VGPR j holds K = 2*(lane>>4)+j
//   B (4x16 f32, 2 VGPRs): lane N = lane&15; VGPR j holds K = 2*(lane>>4)+j
//   C (16x16 f32, 8 VGPRs): VGPR v holds M = v + 8*(lane>>4), N = lane&15
// ---------------------------------------------------------------------------
__global__ void __launch_bounds__(256)
seg_mean_wmma_kernel(const float* __restrict__ x,
                     const float* __restrict__ pe,
                     const int* __restrict__ segstart,
                     const int* __restrict__ segcnt,
                     float* __restrict__ out) {
    const int b    = blockIdx.x;
    const int st   = blockIdx.y;
    const int tid  = threadIdx.x;
    const int wave = tid >> 5;                   // wave32: 8 waves per block
    const int lane = tid & 31;
    const int s0   = st * 16;
    const int d0   = wave * 16;
    const int hi   = lane >> 4;                  // lane half selects K pair
    const int row  = lane & 15;                  // M (for A) / N (for B,C)

    __shared__ int   sh_start[16];
    __shared__ int   sh_cnt[16];
    __shared__ float sh_recip[16];
    if (tid < 16) {
        int stt = segstart[b * MAXSEG + s0 + tid];
        int cc  = segcnt  [b * MAXSEG + s0 + tid];
        sh_start[tid] = stt;
        sh_cnt[tid]   = cc;
        sh_recip[tid] = (cc > 0) ? (1.0f / (float)cc) : 0.0f;
    }
    __syncthreads();

    // token range covering all 16 segments of this tile (monotone bounds)
    const int tlo = sh_start[0];
    const int thi = sh_start[15] + sh_cnt[15];
    // this lane's A-row segment bounds
    const int mst  = sh_start[row];
    const unsigned ucnt = (unsigned)sh_cnt[row]; // membership: (t-mst) u< ucnt

    const float* xb = x + (size_t)b * LL * DD + d0 + row;   // row = N for B

    v8f acc = {};
    int t0 = tlo;

    // ---- main loop: 16 tokens / iter, no clamping, 8 loads in one clause ----
    {
        const float* p = xb + (size_t)(tlo + hi * 2) * DD;  // lane's K=0 token
        int trel = tlo + hi * 2 - mst;                      // t - mst
        for (; t0 + 16 <= thi; t0 += 16, p += 16 * DD, trel += 16) {
            // 8 immediate-offset loads (rows 0,1,4,5,8,9,12,13) -> one s_clause
            float b0 = p[0 * DD],  b1 = p[1 * DD];          // chunk 0
            float b2 = p[4 * DD],  b3 = p[5 * DD];          // chunk 1
            float b4 = p[8 * DD],  b5 = p[9 * DD];          // chunk 2
            float b6 = p[12 * DD], b7 = p[13 * DD];         // chunk 3
            __builtin_prefetch(p + 16 * DD, 0, 1);          // 8 KB ahead

            v2f A0, A1, A2, A3;
            A0.x = ((unsigned)(trel + 0)  < ucnt) ? 1.0f : 0.0f;
            A0.y = ((unsigned)(trel + 1)  < ucnt) ? 1.0f : 0.0f;
            A1.x = ((unsigned)(trel + 4)  < ucnt) ? 1.0f : 0.0f;
            A1.y = ((unsigned)(trel + 5)  < ucnt) ? 1.0f : 0.0f;
            A2.x = ((unsigned)(trel + 8)  < ucnt) ? 1.0f : 0.0f;
            A2.y = ((unsigned)(trel + 9)  < ucnt) ? 1.0f : 0.0f;
            A3.x = ((unsigned)(trel + 12) < ucnt) ? 1.0f : 0.0f;
            A3.y = ((unsigned)(trel + 13) < ucnt) ? 1.0f : 0.0f;

            v2f B0; B0.x = b0; B0.y = b1;
            v2f B1; B1.x = b2; B1.y = b3;
            v2f B2; B2.x = b4; B2.y = b5;
            v2f B3; B3.x = b6; B3.y = b7;

            acc = __builtin_amdgcn_wmma_f32_16x16x4_f32(false, A0, false, B0,
                                                        (short)0, acc, false, false);
            acc = __builtin_amdgcn_wmma_f32_16x16x4_f32(false, A1, false, B1,
                                                        (short)0, acc, false, false);
            acc = __builtin_amdgcn_wmma_f32_16x16x4_f32(false, A2, false, B2,
                                                        (short)0, acc, false, false);
            acc = __builtin_amdgcn_wmma_f32_16x16x4_f32(false, A3, false, B3,
                                                        (short)0, acc, false, false);
        }
    }

    // ---- clamped tail: < 16 tokens remain, 4-token steps ----
    for (; t0 < thi; t0 += 4) {
        const int tA = t0 + hi * 2;              // K for VGPR 0
        const int tB = tA + 1;                   // K for VGPR 1

        v2f A, Bm;
        A.x = ((unsigned)(tA - mst) < ucnt) ? 1.0f : 0.0f;
        A.y = ((unsigned)(tB - mst) < ucnt) ? 1.0f : 0.0f;

        const int ta = (tA < LL) ? tA : (LL - 1);    // clamp; A==0 when OOB
        const int tb = (tB < LL) ? tB : (LL - 1);
        Bm.x = xb[(size_t)ta * DD];
        Bm.y = xb[(size_t)tb * DD];

        acc = __builtin_amdgcn_wmma_f32_16x16x4_f32(false, A, false, Bm,
                                                    (short)0, acc, false, false);
    }

    // finalize: mean + positional encoding, store [16 segs x 16 dims]
    float*       op = out + ((size_t)b * MAXSEG + s0) * DD + d0 + row;
    const float* pp = pe  + (size_t)s0 * DD + d0 + row;
    #pragma unroll
    for (int v = 0; v < 8; ++v) {
        const int mrow = v + hi * 8;             // segment row within tile
        const float val = acc[v] * sh_recip[mrow] + pp[(size_t)mrow * DD];
        op[(size_t)mrow * DD] = val;
    }
}

extern "C" void kernel_launch(void* const* d_in, const int* in_sizes, int n_in,
                              void* d_out, int out_size, void* d_ws, size_t ws_size,
                              hipStream_t stream) {
    const float* x    = (const float*)d_in[0];   // [64, 4096, 128] f32
    const int*   mask = (const int*)  d_in[1];   // [64, 4096] i32
    const float* pe   = (const float*)d_in[2];   // [500, 128] f32
    // d_in[3] = max_ingr_num scalar (256), baked into MAXSEG

    int* segstart = (int*)d_ws;                  // [64, 256]
    int* segcnt   = segstart + BB * MAXSEG;      // [64, 256]

    seg_scan_kernel<<<dim3(BB), dim3(256), 0, stream>>>(mask, segstart, segcnt);

    dim3 grid(BB, MAXSEG / 16);                  // 64 x 16 blocks
    seg_mean_wmma_kernel<<<grid, dim3(256), 0, stream>>>(
        x, pe, segstart, segcnt, (float*)d_out);
}